// DCB_88579405512834
// MI455X (gfx1250) — compile-verified
//
#include <hip/hip_runtime.h>

typedef __attribute__((ext_vector_type(2))) float v2f;
typedef __attribute__((ext_vector_type(8))) float v8f;

#define HID 64
#define OUTC 384   // 6 * 64 columns in x_cat

// ---------------- degree / norm precompute ----------------
__global__ void k_deg_init(float* deg, int n) {
    int i = blockIdx.x * blockDim.x + threadIdx.x;
    if (i < n) deg[i] = 1.0f;   // self-loop contributes 1
}

__global__ void k_deg_edges(const long long* ei, float* deg, int E) {
    int e = blockIdx.x * blockDim.x + threadIdx.x;
    if (e < E) {
        int dst = (int)ei[(size_t)E + e];   // edge_index[1][e]
        atomicAdd(&deg[dst], 1.0f);
    }
}

__global__ void k_dinv(float* deg, int n) {
    int i = blockIdx.x * blockDim.x + threadIdx.x;
    if (i < n) deg[i] = rsqrtf(deg[i]);     // deg >= 1 always
}

__global__ void k_copy_x(const float* x, float* xcat, int total) {
    int idx = blockIdx.x * blockDim.x + threadIdx.x;
    if (idx < total) {
        int row = idx >> 6, f = idx & 63;
        xcat[(size_t)row * OUTC + f] = x[idx];
    }
}

__global__ void k_zero(float* p, int total) {
    int idx = blockIdx.x * blockDim.x + threadIdx.x;
    if (idx < total) p[idx] = 0.0f;
}

// ---------------- WMMA GEMM: h = xcat[:, :fan_in] @ W  ([N,fan_in] x [fan_in,64]) ----------------
// Block = 256 threads = 8 waves (wave32). Block computes a 32-row x 64-col strip.
// Wave w: tile_row = w>>2, tile_col = w&3, each wave one 16x16 f32 tile via
// v_wmma_f32_16x16x4_f32 over the K dimension in LDS-staged chunks of 64.
__global__ __launch_bounds__(256) void k_gemm_wmma(const float* __restrict__ xcat,
                                                   const float* __restrict__ W,
                                                   float* __restrict__ h,
                                                   int fan_in, int n) {
    __shared__ float ldsA[32 * 68];   // 32 rows x 64 K, padded to 68 (bank-conflict free)
    __shared__ float ldsB[64 * 64];   // 64 K x 64 cols

    const int t    = threadIdx.x;
    const int lane = t & 31;
    const int wave = t >> 5;
    const int tr   = wave >> 2;       // 0..1
    const int tc   = wave & 3;        // 0..3
    const int rowBase = blockIdx.x * 32;
    const int m     = lane & 15;             // M (A) / N (B) index within tile
    const int khalf = (lane >> 4) << 1;      // K sub-offset: lanes 0-15 -> 0, 16-31 -> 2

    v8f acc = {};

    for (int kc = 0; kc < fan_in; kc += 64) {
        // stage A chunk: rows [rowBase, rowBase+32) x K [kc, kc+64)
        for (int idx = t; idx < 32 * 16; idx += 256) {
            int r = idx >> 4, c4 = (idx & 15) << 2;
            int gr = rowBase + r; if (gr >= n) gr = n - 1;
            const float4 v = *reinterpret_cast<const float4*>(
                &xcat[(size_t)gr * OUTC + kc + c4]);
            float* d = &ldsA[r * 68 + c4];
            d[0] = v.x; d[1] = v.y; d[2] = v.z; d[3] = v.w;
        }
        // stage B chunk: K [kc, kc+64) x 64 cols
        for (int idx = t; idx < 64 * 16; idx += 256) {
            int r = idx >> 4, c4 = (idx & 15) << 2;
            *reinterpret_cast<float4*>(&ldsB[r * 64 + c4]) =
                *reinterpret_cast<const float4*>(&W[(size_t)(kc + r) * HID + c4]);
        }
        __syncthreads();

#pragma unroll
        for (int k = 0; k < 64; k += 4) {
            const int kb = k + khalf;
            v2f a, b;
            // A 16x4 f32 layout: lanes 0-15 hold M=0..15 {K=k,k+1}; lanes 16-31 {K=k+2,k+3}
            a.x = ldsA[(tr * 16 + m) * 68 + kb];
            a.y = ldsA[(tr * 16 + m) * 68 + kb + 1];
            // B 4x16 f32: mirrored striping over lanes / K halves
            b.x = ldsB[kb * 64 + tc * 16 + m];
            b.y = ldsB[(kb + 1) * 64 + tc * 16 + m];
            acc = __builtin_amdgcn_wmma_f32_16x16x4_f32(
                false, a, false, b, (short)0, acc, false, false);
        }
        __syncthreads();
    }

    // C/D layout: VGPR v -> M = v (lanes 0-15) / v+8 (lanes 16-31); N = lane&15
    const int col   = tc * 16 + m;
    const int rhalf = (lane >> 4) << 3;   // 0 or 8
#pragma unroll
    for (int v = 0; v < 8; ++v) {
        int row = rowBase + tr * 16 + rhalf + v;
        if (row < n) h[(size_t)row * HID + col] = acc[v];
    }
}

// ---------------- edge gather + atomic scatter (wave per edge) ----------------
__global__ __launch_bounds__(256) void k_scatter(const long long* __restrict__ ei,
                                                 const float* __restrict__ dinv,
                                                 const float* __restrict__ h,
                                                 float* __restrict__ agg, int E) {
    int e = blockIdx.x * 8 + (threadIdx.x >> 5);
    if (e >= E) return;
    const int lane = threadIdx.x & 31;
    const int src = (int)ei[e];
    const int dst = (int)ei[(size_t)E + e];
    const float nrm = dinv[src] * dinv[dst];
    const size_t sb = (size_t)src * HID, db = (size_t)dst * HID;
    atomicAdd(&agg[db + lane],      h[sb + lane]      * nrm);
    atomicAdd(&agg[db + lane + 32], h[sb + lane + 32] * nrm);
}

// ---------------- epilogue: self-loop + bias + residual + relu ----------------
__global__ void k_finalize(const float* __restrict__ h, const float* __restrict__ agg,
                           const float* __restrict__ dinv, const float* __restrict__ b,
                           float* __restrict__ xcat, int n, int layer) {
    int idx = blockIdx.x * blockDim.x + threadIdx.x;
    if (idx >= n * HID) return;
    const int row = idx >> 6, f = idx & 63;
    const float di = dinv[row];
    const float v = xcat[(size_t)row * OUTC + (layer - 1) * HID + f]  // residual x_prev
                  + agg[idx]
                  + h[idx] * di * di                                   // self-loop edge
                  + b[f];
    xcat[(size_t)row * OUTC + layer * HID + f] = fmaxf(v, 0.0f);
}

extern "C" void kernel_launch(void* const* d_in, const int* in_sizes, int n_in,
                              void* d_out, int out_size, void* d_ws, size_t ws_size,
                              hipStream_t stream) {
    const float*     x  = (const float*)d_in[0];
    const long long* ei = (const long long*)d_in[1];   // int64 edge_index, shape [2,E]
    const int n = in_sizes[0] / HID;
    const int E = in_sizes[1] / 2;

    float* xcat = (float*)d_out;                 // [n, 384]
    float* dinv = (float*)d_ws;                  // [n]
    float* h    = dinv + n;                      // [n, 64]
    float* agg  = h + (size_t)n * HID;           // [n, 64]

    const int nh = n * HID;
    dim3 blk(256);

    k_deg_init <<<(n + 255) / 256, blk, 0, stream>>>(dinv, n);
    k_deg_edges<<<(E + 255) / 256, blk, 0, stream>>>(ei, dinv, E);
    k_dinv     <<<(n + 255) / 256, blk, 0, stream>>>(dinv, n);
    k_copy_x   <<<(nh + 255) / 256, blk, 0, stream>>>(x, xcat, nh);

    for (int layer = 1; layer <= 5; ++layer) {
        const int fan_in = layer * HID;
        const float* W = (const float*)d_in[2 * layer];      // W_layer
        const float* b = (const float*)d_in[2 * layer + 1];  // b_layer

        k_zero<<<(nh + 255) / 256, blk, 0, stream>>>(agg, nh);
        k_gemm_wmma<<<(n + 31) / 32, blk, 0, stream>>>(xcat, W, h, fan_in, n);
        k_scatter<<<(E + 7) / 8, blk, 0, stream>>>(ei, dinv, h, agg, E);
        k_finalize<<<(nh + 255) / 256, blk, 0, stream>>>(h, agg, dinv, b, xcat, n, layer);
    }
}